// LabelLoss_59115929862965
// MI455X (gfx1250) — compile-verified
//
#include <hip/hip_runtime.h>

// LabelLoss: memory-bound masked reduction.
//   out[b] = sum_o [ valid(b,o) * sum_{c<7} (pred[b,o,c]-gt[b,o,c])^2 ]
//   valid  = 0 < gt[...,9] < m  &&  0 < gt[...,10] < n   (m=n=128 from heatmap shape)
// pred: [B, O, 16] f32, gt: [B, O, 11] f32, heatmap only supplies (m, n).

typedef float v2f __attribute__((ext_vector_type(2)));
typedef float v4f __attribute__((ext_vector_type(4)));
typedef float v8f __attribute__((ext_vector_type(8)));

#define THREADS 512  // 16 wave32s per block

__global__ __launch_bounds__(THREADS)
void label_loss_kernel(const float* __restrict__ pred,
                       const float* __restrict__ gt,
                       float* __restrict__ out,
                       int O, float m, float n) {
  const int b   = blockIdx.x;
  const int tid = threadIdx.x;

  const float* __restrict__ predB = pred + (size_t)b * (size_t)O * 16u;
  const float* __restrict__ gtB   = gt   + (size_t)b * (size_t)O * 11u;

  float acc = 0.0f;

  for (int o = tid; o < O; o += THREADS) {
    const float* pr = predB + (size_t)o * 16u;
    const float* gr = gtB   + (size_t)o * 11u;

    // Stream-ahead prefetch of the next strip (gfx1250 global_prefetch_b8).
    if (o + THREADS < O) {
      __builtin_prefetch(pr + (size_t)THREADS * 16u, 0, 0);
      __builtin_prefetch(gr + (size_t)THREADS * 11u, 0, 0);
    }

    // pred row: channels 0..7 via two 128-bit non-temporal loads (16B aligned:
    // row stride is 64B). Working set (216 MB) exceeds L2 (192 MB) -> NT.
    v4f p0 = __builtin_nontemporal_load((const v4f*)pr);
    v4f p1 = __builtin_nontemporal_load((const v4f*)(pr + 4));

    // gt row stride is 44B (not 16B aligned) -> dword NT loads.
    float g0 = __builtin_nontemporal_load(gr + 0);
    float g1 = __builtin_nontemporal_load(gr + 1);
    float g2 = __builtin_nontemporal_load(gr + 2);
    float g3 = __builtin_nontemporal_load(gr + 3);
    float g4 = __builtin_nontemporal_load(gr + 4);
    float g5 = __builtin_nontemporal_load(gr + 5);
    float g6 = __builtin_nontemporal_load(gr + 6);
    float cx = __builtin_nontemporal_load(gr + 9);
    float cy = __builtin_nontemporal_load(gr + 10);

    float d0 = p0.x - g0;
    float d1 = p0.y - g1;
    float d2 = p0.z - g2;
    float d3 = p0.w - g3;
    float d4 = p1.x - g4;
    float d5 = p1.y - g5;
    float d6 = p1.z - g6;

    float s = d0 * d0;
    s = fmaf(d1, d1, s);
    s = fmaf(d2, d2, s);
    s = fmaf(d3, d3, s);
    s = fmaf(d4, d4, s);
    s = fmaf(d5, d5, s);
    s = fmaf(d6, d6, s);

    bool valid = (cx > 0.0f) && (cy > 0.0f) && (cx < m) && (cy < n);
    acc += valid ? s : 0.0f;
  }

  // Block reduction: 512 partials -> 32 via LDS tree.
  __shared__ float sdata[THREADS];
  sdata[tid] = acc;
  __syncthreads();
#pragma unroll
  for (int s = THREADS / 2; s >= 32; s >>= 1) {
    if (tid < s) sdata[tid] += sdata[tid + s];
    __syncthreads();
  }

  // Final 32 -> 1 with one exact f32 WMMA (wave 0 only; EXEC all-1s in-wave).
  // A layout (16x4 f32): lane L<16 -> A[M=L,K=0]=p_L (VGPR0), A[M=L,K=1]=0;
  //                      lane L>=16 -> A[M=L-16,K=2]=p_L,     A[M=L-16,K=3]=0.
  // B = all ones (value is layout-invariant), so D[m,n] = p_m + p_{m+16}.
  // Lane L<16 holds D[0..7, L]; lane L>=16 holds D[8..15, L-16].
  if (tid < 32) {
    v2f a;
    a.x = sdata[tid];
    a.y = 0.0f;
    v2f bo;
    bo.x = 1.0f;
    bo.y = 1.0f;
    v8f c = {};
    v8f d = __builtin_amdgcn_wmma_f32_16x16x4_f32(
        /*neg_a=*/false, a, /*neg_b=*/false, bo,
        /*c_mod=*/(short)0, c, /*reuse_a=*/false, /*reuse_b=*/false);
    float s8 = ((d[0] + d[1]) + (d[2] + d[3])) + ((d[4] + d[5]) + (d[6] + d[7]));
    sdata[tid] = s8;  // lane 0: sum of rows 0..7; lane 16: sum of rows 8..15
  }
  __syncthreads();

  if (tid == 0) out[b] = sdata[0] + sdata[16];
}

extern "C" void kernel_launch(void* const* d_in, const int* in_sizes, int n_in,
                              void* d_out, int out_size, void* d_ws, size_t ws_size,
                              hipStream_t stream) {
  const float* pred = (const float*)d_in[0];
  const float* gt   = (const float*)d_in[1];
  // d_in[2] (heatmap) supplies only its spatial dims; data never read.
  float* out = (float*)d_out;

  const int B = out_size;                         // 256
  const int O = (B > 0) ? in_sizes[0] / (B * 16) : 0;  // 8192

  // Recover (m, n) from heatmap element count: B*1*m*n, square per reference.
  int mn = (B > 0 && n_in > 2) ? in_sizes[2] / B : 128 * 128;
  int mi = 1;
  while ((long long)(mi + 1) * (mi + 1) <= (long long)mn) ++mi;  // isqrt
  const float mf = (float)mi;
  const float nf = (float)(mi > 0 ? mn / mi : mn);

  label_loss_kernel<<<B, THREADS, 0, stream>>>(pred, gt, out, O, mf, nf);
}